// multi_head_attention_72430328481544
// MI455X (gfx1250) — compile-verified
//
#include <hip/hip_runtime.h>
#include <hip/hip_bf16.h>

#define D_MODEL 1024
#define N_HEADS 16
#define D_HEAD  64
#define BATCH   2
#define SEQ     2048

// padded LDS row stride (floats): 2052 = 32*64 + 4 -> row m starts at bank 4m,
// and 2052*4 B = 8208 B keeps rows 16-byte aligned for b128 access
#define LDS_STRIDE 2052

typedef __attribute__((ext_vector_type(2))) float v2f;
typedef __attribute__((ext_vector_type(4))) float v4f;
typedef __attribute__((ext_vector_type(8))) float v8f;

__device__ __forceinline__ v8f wmma_f32_k4(v2f a, v2f b, v8f c) {
  // D = A(16x4,f32) * B(4x16,f32) + C(16x16,f32)
  return __builtin_amdgcn_wmma_f32_16x16x4_f32(
      false, a, false, b, (short)0, c, false, false);
}

// ---------------------------------------------------------------------------
// Kernel 1: Y[B,H,S,d] = head_split(X[BS,D] @ W[D,D] + bias[D])
// one wave -> one 16x16 output tile, K-loop over D in steps of 4
// ---------------------------------------------------------------------------
__global__ void __launch_bounds__(32) qkv_proj_kernel(
    const float* __restrict__ X, const float* __restrict__ W,
    const float* __restrict__ bias, float* __restrict__ Y) {
  const int lane = threadIdx.x;
  const int n    = lane & 15;          // output col within tile / A row
  const int koff = (lane >> 4) * 2;    // K sub-offset per half-wave
  const int col0 = blockIdx.x * 16;    // over D_MODEL
  const int row0 = blockIdx.y * 16;    // over B*S

  const float* xrow = X + (size_t)(row0 + n) * D_MODEL;        // A row m = lane&15
  const float* wcol = W + col0 + n;                            // B col n

  v8f c = {};
  for (int kk = 0; kk < D_MODEL; kk += 4) {
    v2f a = *(const v2f*)(xrow + kk + koff);                   // A[m][kk+koff..+1]
    v2f b;
    b.x = wcol[(size_t)(kk + koff) * D_MODEL];                 // B[kk+koff][n]
    b.y = wcol[(size_t)(kk + koff + 1) * D_MODEL];
    c = wmma_f32_k4(a, b, c);
  }

  const int mbase = (lane >> 4) * 8;
  const int gcol  = col0 + n;
  const int h     = gcol >> 6;     // / D_HEAD
  const int dd    = gcol & 63;
  const float bv  = bias[gcol];
#pragma unroll
  for (int r = 0; r < 8; ++r) {
    const int grow = row0 + mbase + r;       // 0..B*S-1
    const int bb = grow >> 11;               // / SEQ
    const int ss = grow & (SEQ - 1);
    Y[(((size_t)bb * N_HEADS + h) * SEQ + ss) * D_HEAD + dd] = c[r] + bv;
  }
}

// ---------------------------------------------------------------------------
// Kernel 2 (fused): per (h,b,qtile) workgroup of 8 waves
//   P1: scores(16x2048) -> LDS   (WMMA, scaled by 1/sqrt(64))
//   P2: row softmax in LDS (wave32 shfl reductions), single b128 global write
//   P3: ctx(16x64) = att(LDS) @ V (WMMA, split over waves, LDS reduction)
// ---------------------------------------------------------------------------
__global__ void __launch_bounds__(256) attention_fused_kernel(
    const float* __restrict__ qh, const float* __restrict__ kh,
    const float* __restrict__ vh, float* __restrict__ att,
    float* __restrict__ ctx) {
  extern __shared__ float smem[];
  float* s_att  = smem;                       // 16 * LDS_STRIDE floats
  float* s_part = smem + 16 * LDS_STRIDE;     // 4 * 256 floats

  const int t     = threadIdx.x;
  const int lane  = t & 31;
  const int w     = t >> 5;                   // wave id 0..7
  const int n     = lane & 15;
  const int koff  = (lane >> 4) * 2;
  const int mbase = (lane >> 4) * 8;

  const int qtile = blockIdx.x;               // 0..127
  const int hb    = blockIdx.y;               // h*B + b
  const int h = hb / BATCH;
  const int b = hb % BATCH;
  const size_t base = ((size_t)b * N_HEADS + h) * SEQ;

  // ---- Phase 1: scores into LDS ----
  // hoist this wave's Q-tile A-fragments (16 WMMA steps * v2f) into registers
  const float* qrow = qh + (base + (size_t)qtile * 16 + n) * D_HEAD;
  v2f a_reg[16];
#pragma unroll
  for (int i = 0; i < 16; ++i)
    a_reg[i] = *(const v2f*)(qrow + 4 * i + koff);

  for (int kt = w * 16; kt < w * 16 + 16; ++kt) {   // 16 ktiles per wave
    const float* krow = kh + (base + (size_t)kt * 16 + n) * D_HEAD;
    v8f c = {};
#pragma unroll
    for (int i = 0; i < 16; ++i) {
      v2f bm = *(const v2f*)(krow + 4 * i + koff);  // B[kk][n] = K[n][kk]
      c = wmma_f32_k4(a_reg[i], bm, c);
    }
#pragma unroll
    for (int r = 0; r < 8; ++r)
      s_att[(mbase + r) * LDS_STRIDE + kt * 16 + n] = c[r] * 0.125f;
  }
  __syncthreads();

  // ---- Phase 2: softmax (2 rows per wave); b128 LDS reads + global writes ----
  for (int rr = 0; rr < 2; ++rr) {
    const int row = w * 2 + rr;
    float* srow = s_att + row * LDS_STRIDE;

    v4f vals[16];                 // lane owns cols [lane*4 + i*128, +4)
    float lmax = -3.4e38f;
#pragma unroll
    for (int i = 0; i < 16; ++i) {
      vals[i] = *(const v4f*)(srow + lane * 4 + i * 128);      // ds_load_b128
      lmax = fmaxf(lmax, fmaxf(fmaxf(vals[i].x, vals[i].y),
                               fmaxf(vals[i].z, vals[i].w)));
    }
#pragma unroll
    for (int off = 16; off > 0; off >>= 1)
      lmax = fmaxf(lmax, __shfl_xor(lmax, off, 32));

    float lsum = 0.f;
#pragma unroll
    for (int i = 0; i < 16; ++i) {
#pragma unroll
      for (int j = 0; j < 4; ++j) {
        vals[i][j] = __expf(vals[i][j] - lmax);
        lsum += vals[i][j];
      }
    }
#pragma unroll
    for (int off = 16; off > 0; off >>= 1)
      lsum += __shfl_xor(lsum, off, 32);
    const float inv = 1.0f / lsum;

    float* grow = att + ((size_t)hb * SEQ + (size_t)qtile * 16 + row) * SEQ;
#pragma unroll
    for (int i = 0; i < 16; ++i) {
      const v4f vv = vals[i] * inv;
      *(v4f*)(srow + lane * 4 + i * 128) = vv;     // for phase 3
      *(v4f*)(grow + lane * 4 + i * 128) = vv;     // global_store_b128 (output)
    }
  }
  __syncthreads();

  // ---- Phase 3: ctx = att(LDS) @ V ----
  const int dtile = w & 3;                    // 4 x 16 output cols
  const int khalf = w >> 2;                   // K split across wave pairs
  const float* vcol   = vh + base * D_HEAD + dtile * 16 + n;   // B col n
  const float* arow_l = s_att + (size_t)n * LDS_STRIDE;        // A row m = lane&15

  v8f c = {};
  const int k0 = khalf * (SEQ / 2);
  for (int kk = k0; kk < k0 + SEQ / 2; kk += 4) {
    v2f a = *(const v2f*)(arow_l + kk + koff);                 // ds_load_b64
    v2f bm;
    bm.x = vcol[(size_t)(kk + koff) * D_HEAD];                 // B[kk][n] = V[kk][n]
    bm.y = vcol[(size_t)(kk + koff + 1) * D_HEAD];
    c = wmma_f32_k4(a, bm, c);
  }

  if (w >= 4) {
#pragma unroll
    for (int r = 0; r < 8; ++r)
      s_part[dtile * 256 + r * 32 + lane] = c[r];
  }
  __syncthreads();
  if (w < 4) {
#pragma unroll
    for (int r = 0; r < 8; ++r)
      c[r] += s_part[dtile * 256 + r * 32 + lane];
    float* out = ctx + ((size_t)b * SEQ + (size_t)qtile * 16) * D_MODEL
                     + h * D_HEAD + dtile * 16 + n;
#pragma unroll
    for (int r = 0; r < 8; ++r)
      out[(size_t)(mbase + r) * D_MODEL] = c[r];
  }
}

// ---------------------------------------------------------------------------
extern "C" void kernel_launch(void* const* d_in, const int* in_sizes, int n_in,
                              void* d_out, int out_size, void* d_ws, size_t ws_size,
                              hipStream_t stream) {
  (void)in_sizes; (void)n_in; (void)out_size; (void)ws_size;

  const size_t BHSd = (size_t)BATCH * N_HEADS * SEQ * D_HEAD;  // 4 Mi floats
  float* ws = (float*)d_ws;
  float* qh = ws;
  float* kh = ws + BHSd;
  float* vh = ws + 2 * BHSd;

  float* ctx = (float*)d_out;                                   // [B,S,D]
  float* att = (float*)d_out + (size_t)BATCH * SEQ * D_MODEL;   // [h*B,S,S]

  const float* query = (const float*)d_in[0];
  const float* key_  = (const float*)d_in[1];
  const float* value = (const float*)d_in[2];
  const float* Wq = (const float*)d_in[3]; const float* bq = (const float*)d_in[4];
  const float* Wk = (const float*)d_in[5]; const float* bk = (const float*)d_in[6];
  const float* Wv = (const float*)d_in[7]; const float* bv = (const float*)d_in[8];

  {
    dim3 grid(D_MODEL / 16, (BATCH * SEQ) / 16, 1);
    qkv_proj_kernel<<<grid, 32, 0, stream>>>(query, Wq, bq, qh);
    qkv_proj_kernel<<<grid, 32, 0, stream>>>(key_,  Wk, bk, kh);
    qkv_proj_kernel<<<grid, 32, 0, stream>>>(value, Wv, bv, vh);
  }
  {
    dim3 grid(SEQ / 16, N_HEADS * BATCH);
    const size_t lds_bytes = (16 * LDS_STRIDE + 4 * 256) * sizeof(float); // ~133 KB
    attention_fused_kernel<<<grid, 256, lds_bytes, stream>>>(qh, kh, vh, att, ctx);
  }
}